// Imputer_9998683865331
// MI455X (gfx1250) — compile-verified
//
#include <hip/hip_runtime.h>

typedef __attribute__((ext_vector_type(2))) float v2f;
typedef __attribute__((ext_vector_type(8))) float v8f;

#define HID0 128

__device__ __forceinline__ float relu_f(float v) { return v > 0.f ? v : 0.f; }

// ---------------- degree / normalization / scatter ----------------

__global__ void k_deg_init(float* __restrict__ deg, int n) {
  int i = blockIdx.x * blockDim.x + threadIdx.x;
  if (i < n) deg[i] = 1.0f;  // self-loop contribution
}

__global__ void k_deg_accum(const int* __restrict__ col, float* __restrict__ deg, int E) {
  int e = blockIdx.x * blockDim.x + threadIdx.x;
  if (e < E) atomicAdd(&deg[col[e]], 1.0f);
}

__global__ void k_rsqrt_inplace(float* __restrict__ deg, int n) {
  int i = blockIdx.x * blockDim.x + threadIdx.x;
  if (i < n) deg[i] = rsqrtf(deg[i]);  // deg >= 1 always (self-loops)
}

// agg[i,:] = h0[i,:] * dis[i]^2   (self-loop edge of the normalized adjacency)
__global__ void k_agg_selfloop(const float* __restrict__ h0, const float* __restrict__ dis,
                               float* __restrict__ agg, int total) {
  int i = blockIdx.x * blockDim.x + threadIdx.x;
  if (i < total) {
    float d = dis[i >> 7];  // /HID0
    agg[i] = h0[i] * d * d;
  }
}

// one wave32 per edge; each lane owns 4 of the 128 channels
__global__ void k_edge_scatter(const int* __restrict__ row, const int* __restrict__ col,
                               const float* __restrict__ h0, const float* __restrict__ dis,
                               float* __restrict__ agg, int E) {
  int t = blockIdx.x * blockDim.x + threadIdx.x;
  int e = t >> 5, lane = t & 31;
  if (e >= E) return;
  int r = row[e], c = col[e];
  float w = dis[r] * dis[c];
  const float4 v = *(reinterpret_cast<const float4*>(h0 + (size_t)r * HID0) + lane);
  float* dst = agg + (size_t)c * HID0 + lane * 4;
  atomicAdd(dst + 0, v.x * w);
  atomicAdd(dst + 1, v.y * w);
  atomicAdd(dst + 2, v.z * w);
  atomicAdd(dst + 3, v.w * w);
}

// ---------------- fp32 WMMA GEMM: C = act(act_in(A + bin) @ W + bout) ----------------
// Block = 256 threads = 8 wave32s. Block tile: 128 rows x N cols.
// LDS: W [K*N] + A-slab [128*K], both fp32 (max 256KB < 320KB/WGP).
// Wave w owns rows [blk*128 + w*16, +16); loops N/16 col tiles, K/4 WMMA steps.
// Uniform fast path when the whole 128-row block is in range (all but last block)
// avoids per-row exec-mask guards.
template <int K, int N, bool RELU_IN, bool RELU_OUT, bool HAS_BOUT>
__global__ void k_gemm(const float* __restrict__ A, const float* __restrict__ bin,
                       const float* __restrict__ W, const float* __restrict__ bout,
                       float* __restrict__ C, int M) {
  extern __shared__ float smem[];
  float* Wl = smem;            // K*N floats
  float* Al = smem + K * N;    // 128*K floats

  const int tid = threadIdx.x;
  const int wv = tid >> 5;
  const int lane = tid & 31;
  const int hl = lane >> 4;    // half-wave select
  const int ll = lane & 15;
  const int R0 = blockIdx.x * 128;
  const bool full = (R0 + 128 <= M);   // wave-uniform

  // stage weights (row-major K x N)
  for (int idx = tid; idx < (K * N) / 4; idx += 256)
    *(reinterpret_cast<float4*>(Wl) + idx) = *(reinterpret_cast<const float4*>(W) + idx);

  // stage A slab (row-major 128 x K), optional fused relu(a + bin[k]); zero-fill tail rows
  if (full) {
    for (int idx = tid; idx < (128 * K) / 4; idx += 256) {
      int r = idx / (K / 4);
      int kq = idx % (K / 4);
      float4 v = *(reinterpret_cast<const float4*>(A + (size_t)(R0 + r) * K) + kq);
      if (RELU_IN) {
        const float4 b = *(reinterpret_cast<const float4*>(bin) + kq);
        v.x = relu_f(v.x + b.x);
        v.y = relu_f(v.y + b.y);
        v.z = relu_f(v.z + b.z);
        v.w = relu_f(v.w + b.w);
      }
      *(reinterpret_cast<float4*>(Al) + idx) = v;
    }
  } else {
    for (int idx = tid; idx < (128 * K) / 4; idx += 256) {
      int r = idx / (K / 4);
      int kq = idx % (K / 4);
      float4 v = make_float4(0.f, 0.f, 0.f, 0.f);
      if (R0 + r < M) {
        v = *(reinterpret_cast<const float4*>(A + (size_t)(R0 + r) * K) + kq);
        if (RELU_IN) {
          const float4 b = *(reinterpret_cast<const float4*>(bin) + kq);
          v.x = relu_f(v.x + b.x);
          v.y = relu_f(v.y + b.y);
          v.z = relu_f(v.z + b.z);
          v.w = relu_f(v.w + b.w);
        }
      }
      *(reinterpret_cast<float4*>(Al) + idx) = v;
    }
  }
  __syncthreads();

  // ISA 7.12.2 layouts (wave32):
  //   A 16x4 frag:  a[j] = A[M=ll][Kidx = k0 + 2*hl + j]
  //   B 4x16 frag:  b[j] = B[Kidx = k0 + 2*hl + j][N = n0 + ll]
  //   C/D 16x16:    c[v] = C[M = v + 8*hl][N = n0 + ll]
  const float* Arow = Al + (size_t)(wv * 16 + ll) * K;
  const int slab0 = R0 + wv * 16;

  for (int nt = 0; nt < N / 16; ++nt) {
    v8f acc = {};
    const float* Wc = Wl + nt * 16 + ll;
#pragma unroll 4
    for (int k0 = 0; k0 < K; k0 += 4) {
      v2f a = *reinterpret_cast<const v2f*>(Arow + k0 + 2 * hl);
      v2f b;
      b[0] = Wc[(k0 + 2 * hl) * N];
      b[1] = Wc[(k0 + 2 * hl + 1) * N];
      acc = __builtin_amdgcn_wmma_f32_16x16x4_f32(false, a, false, b, (short)0, acc,
                                                  false, false);
    }
    const float bb = HAS_BOUT ? bout[nt * 16 + ll] : 0.0f;
    float* Cp = C + (size_t)(slab0 + 8 * hl) * N + nt * 16 + ll;
    if (full) {
#pragma unroll
      for (int v = 0; v < 8; ++v) {
        float o = acc[v] + bb;
        if (RELU_OUT) o = relu_f(o);
        Cp[(size_t)v * N] = o;
      }
    } else {
#pragma unroll
      for (int v = 0; v < 8; ++v) {
        if (slab0 + 8 * hl + v < M) {
          float o = acc[v] + bb;
          if (RELU_OUT) o = relu_f(o);
          Cp[(size_t)v * N] = o;
        }
      }
    }
  }
}

// ---------------- launcher ----------------

extern "C" void kernel_launch(void* const* d_in, const int* in_sizes, int n_in,
                              void* d_out, int out_size, void* d_ws, size_t ws_size,
                              hipStream_t stream) {
  const float* x      = (const float*)d_in[0];
  const int*   ei     = (const int*)d_in[1];
  const float* W_conv = (const float*)d_in[2];
  const float* b_conv = (const float*)d_in[3];
  const float* W_lin  = (const float*)d_in[4];
  const float* b_lin  = (const float*)d_in[5];
  const float* W_dec1 = (const float*)d_in[6];
  const float* b_dec1 = (const float*)d_in[7];
  const float* W_dec2 = (const float*)d_in[8];
  const float* b_dec2 = (const float*)d_in[9];
  float* out = (float*)d_out;

  const int N = in_sizes[0] / 256;   // nodes
  const int E = in_sizes[1] / 2;     // edges
  const int* erow = ei;              // edge_index[0]
  const int* ecol = ei + E;          // edge_index[1]

  float* ws  = (float*)d_ws;
  float* h0  = ws;                          // N*128
  float* agg = ws + (size_t)N * HID0;       // N*128
  float* deg = ws + (size_t)2 * N * HID0;   // N (becomes dis in-place)
  float* z   = h0;                          // reuse: h0 dead after scatter
  float* d1  = agg;                         // reuse: agg dead after stage z

  const int MB = (N + 127) / 128;

  // 1) h0 = x @ W_conv   (no bias, per GCNConv linear)
  k_gemm<256, 128, false, false, false>
      <<<MB, 256, (256 * 128 + 128 * 256) * sizeof(float), stream>>>(
          x, nullptr, W_conv, nullptr, h0, N);

  // 2) degrees -> dis = rsqrt(deg)
  k_deg_init<<<(N + 255) / 256, 256, 0, stream>>>(deg, N);
  k_deg_accum<<<(E + 255) / 256, 256, 0, stream>>>(ecol, deg, E);
  k_rsqrt_inplace<<<(N + 255) / 256, 256, 0, stream>>>(deg, N);

  // 3) agg = self-loop term + edge scatter-add
  k_agg_selfloop<<<((N * HID0) + 255) / 256, 256, 0, stream>>>(h0, deg, agg, N * HID0);
  k_edge_scatter<<<(E + 7) / 8, 256, 0, stream>>>(erow, ecol, h0, deg, agg, E);

  // 4) z = relu(agg + b_conv) @ W_lin + b_lin
  k_gemm<128, 64, true, false, true>
      <<<MB, 256, (128 * 64 + 128 * 128) * sizeof(float), stream>>>(
          agg, b_conv, W_lin, b_lin, z, N);

  // 5) d = relu(z @ W_dec1 + b_dec1)
  k_gemm<64, 128, false, true, true>
      <<<MB, 256, (64 * 128 + 128 * 64) * sizeof(float), stream>>>(
          z, nullptr, W_dec1, b_dec1, d1, N);

  // 6) out = d @ W_dec2 + b_dec2
  k_gemm<128, 256, false, false, true>
      <<<MB, 256, (128 * 256 + 128 * 128) * sizeof(float), stream>>>(
          d1, nullptr, W_dec2, b_dec2, out, N);
}